// CompetenceAssessor_59485297049951
// MI455X (gfx1250) — compile-verified
//
#include <hip/hip_runtime.h>
#include <hip/hip_bf16.h>
#include <math.h>

// CDNA5 / gfx1250: wave32, WMMA 16x16x4 f32
typedef __attribute__((ext_vector_type(2))) float v2f;
typedef __attribute__((ext_vector_type(8))) float v8f;

#define HIST   100
#define RECENT 20
#define HIDDEN 128
#define H2DIM  32
#define BLOCK  256

__global__ __launch_bounds__(BLOCK)
void competence_assessor_kernel(const float* __restrict__ hist,
                                const float* __restrict__ W1, const float* __restrict__ b1,
                                const float* __restrict__ W2, const float* __restrict__ b2,
                                const float* __restrict__ W3, const float* __restrict__ b3,
                                float* __restrict__ out, int batch)
{
    // ---- LDS: weights (shared by block) + per-wave h2 staging tile ----
    __shared__ float4 sW1P[HIDDEN];            // packed {W1[0][j], W1[1][j], W1[2][j], b1[j]}
    __shared__ float  sW2[HIDDEN * H2DIM];     // W2[k][n] row-major (128 x 32)
    __shared__ float  sB2[H2DIM];
    __shared__ float  sW3[H2DIM];
    __shared__ float  sB3[1];
    __shared__ float  sH2[BLOCK / 32][32][33]; // padded to 33 to avoid bank conflicts

    const int tid = threadIdx.x;

    for (int j = tid; j < HIDDEN; j += BLOCK)
        sW1P[j] = make_float4(W1[j], W1[HIDDEN + j], W1[2 * HIDDEN + j], b1[j]);
    for (int i = tid; i < HIDDEN * H2DIM; i += BLOCK) sW2[i] = W2[i];
    if (tid < H2DIM) { sB2[tid] = b2[tid]; sW3[tid] = W3[tid]; }
    if (tid == 0)    sB3[0] = b3[0];
    __syncthreads();

    const int lane = tid & 31;
    const int wave = tid >> 5;
    const bool lo  = lane < 16;
    const int n16  = lane & 15;

    const int r0 = blockIdx.x * BLOCK + tid;        // this thread's row
    const int r  = (r0 < batch) ? r0 : (batch - 1); // clamp: keep EXEC full for WMMA

    // ================= feature extraction (HBM-bound streaming) =================
    // Row = 100 x (success, difficulty) fp32 pairs = 800B, 16B aligned -> float4.
    const float4* __restrict__ p = (const float4*)(hist + (size_t)r * (2 * HIST));
    float sum = 0.f, early = 0.f, recent = 0.f, maxd = 0.f;

    #pragma unroll
    for (int i = 0; i < 10; ++i) {                  // t = 0..19 (early window)
        float4 v = p[i];
        sum += v.x + v.z; early += v.x + v.z;
        if (v.x > 0.5f) maxd = fmaxf(maxd, v.y);
        if (v.z > 0.5f) maxd = fmaxf(maxd, v.w);
    }
    #pragma unroll 10
    for (int i = 10; i < 40; ++i) {                 // t = 20..79
        float4 v = p[i];
        sum += v.x + v.z;
        if (v.x > 0.5f) maxd = fmaxf(maxd, v.y);
        if (v.z > 0.5f) maxd = fmaxf(maxd, v.w);
    }
    #pragma unroll
    for (int i = 40; i < 50; ++i) {                 // t = 80..99 (recent window)
        float4 v = p[i];
        sum += v.x + v.z; recent += v.x + v.z;
        if (v.x > 0.5f) maxd = fmaxf(maxd, v.y);
        if (v.z > 0.5f) maxd = fmaxf(maxd, v.w);
    }

    const float f0 = sum * (1.0f / HIST);
    const float f1 = (recent - early) * (1.0f / RECENT);
    const float f2 = maxd;

    // Exchange features with the partner lane (lane ^ 16) ONCE, so the K-loop
    // needs no cross-lane ops at all. fa* = features for the row whose h1 values
    // this lane supplies in the rows-0..15 A tile; fb* = rows-16..31 A tile.
    const float p0 = __shfl_xor(f0, 16, 32);
    const float p1 = __shfl_xor(f1, 16, 32);
    const float p2 = __shfl_xor(f2, 16, 32);
    const float fa0 = lo ? f0 : p0, fa1 = lo ? f1 : p1, fa2 = lo ? f2 : p2;
    const float fb0 = lo ? p0 : f0, fb1 = lo ? p1 : f1, fb2 = lo ? p2 : f2;

    // ================= layer 2 accumulators (16x16 f32 D tiles) =================
    // acc[rowtile][coltile]; C initialized with b2 (column-only -> broadcast per lane)
    const float c0 = sB2[n16];
    const float c1 = sB2[16 + n16];
    v8f acc00 = {c0, c0, c0, c0, c0, c0, c0, c0};
    v8f acc01 = {c1, c1, c1, c1, c1, c1, c1, c1};
    v8f acc10 = acc00;
    v8f acc11 = acc01;

    // ================= fused layer1 (VALU) + layer2 (WMMA f32 16x16x4) ==========
    // A-layout for f32 16x4: lanes 0-15 hold (K0,K1) of row=lane; lanes 16-31 hold
    // (K2,K3) of row=lane-16.  Each lane computes exactly the h1 values its slot
    // needs: hidden indices jj,jj+1 with jj = 4*kk + (lo ? 0 : 2), from feature
    // set fa (row tile 0) and fb (row tile 1).  No shuffles in the loop.
    #pragma unroll 8
    for (int kk = 0; kk < HIDDEN / 4; ++kk) {
        const int jj = kk * 4 + (lo ? 0 : 2);
        const float4 wa = sW1P[jj];        // {w0,w1,w2,b} for hidden unit jj
        const float4 wb = sW1P[jj + 1];    // {w0,w1,w2,b} for hidden unit jj+1

        v2f a_r0, a_r1;
        a_r0.x = fmaxf(wa.w + fa0 * wa.x + fa1 * wa.y + fa2 * wa.z, 0.f);
        a_r0.y = fmaxf(wb.w + fa0 * wb.x + fa1 * wb.y + fa2 * wb.z, 0.f);
        a_r1.x = fmaxf(wa.w + fb0 * wa.x + fb1 * wa.y + fb2 * wa.z, 0.f);
        a_r1.y = fmaxf(wb.w + fb0 * wb.x + fb1 * wb.y + fb2 * wb.z, 0.f);

        // B (4x16 f32): VGPR0 = K row jj (lanes 0-15 get K0-row, lanes 16-31 get
        // K2-row), VGPR1 = K row jj+1.  N = 16*coltile + (lane&15).
        v2f b_c0, b_c1;
        b_c0.x = sW2[jj * H2DIM + n16];
        b_c0.y = sW2[(jj + 1) * H2DIM + n16];
        b_c1.x = sW2[jj * H2DIM + 16 + n16];
        b_c1.y = sW2[(jj + 1) * H2DIM + 16 + n16];

        acc00 = __builtin_amdgcn_wmma_f32_16x16x4_f32(false, a_r0, false, b_c0, (short)0, acc00, false, false);
        acc01 = __builtin_amdgcn_wmma_f32_16x16x4_f32(false, a_r0, false, b_c1, (short)0, acc01, false, false);
        acc10 = __builtin_amdgcn_wmma_f32_16x16x4_f32(false, a_r1, false, b_c0, (short)0, acc10, false, false);
        acc11 = __builtin_amdgcn_wmma_f32_16x16x4_f32(false, a_r1, false, b_c1, (short)0, acc11, false, false);
    }

    // ================= layer 3 (32 -> 1) + sigmoid ==============================
    // D layout: lane n holds column (16*ct + n&15); VGPR v = row (v + 8*(lane>=16)).
    const int rbase = lo ? 0 : 8;
    #pragma unroll
    for (int v = 0; v < 8; ++v) {
        sH2[wave][rbase + v][n16]           = fmaxf(acc00[v], 0.f);
        sH2[wave][rbase + v][16 + n16]      = fmaxf(acc01[v], 0.f);
        sH2[wave][16 + rbase + v][n16]      = fmaxf(acc10[v], 0.f);
        sH2[wave][16 + rbase + v][16 + n16] = fmaxf(acc11[v], 0.f);
    }
    __syncthreads();

    float o = sB3[0];
    #pragma unroll
    for (int c = 0; c < H2DIM; ++c) o += sH2[wave][lane][c] * sW3[c];
    o = 1.0f / (1.0f + __expf(-o));

    if (r0 < batch) out[r0] = o;
}

extern "C" void kernel_launch(void* const* d_in, const int* in_sizes, int n_in,
                              void* d_out, int out_size, void* d_ws, size_t ws_size,
                              hipStream_t stream) {
    const float* hist = (const float*)d_in[0];
    const float* W1   = (const float*)d_in[1];
    const float* b1   = (const float*)d_in[2];
    const float* W2   = (const float*)d_in[3];
    const float* b2   = (const float*)d_in[4];
    const float* W3   = (const float*)d_in[5];
    const float* b3   = (const float*)d_in[6];
    float* out = (float*)d_out;

    const int batch  = in_sizes[0] / (2 * HIST);
    const int blocks = (batch + BLOCK - 1) / BLOCK;
    competence_assessor_kernel<<<blocks, BLOCK, 0, stream>>>(hist, W1, b1, W2, b2, W3, b3, out, batch);
}